// HeteroRGCN_50388556317054
// MI455X (gfx1250) — compile-verified
//
#include <hip/hip_runtime.h>

typedef float v2f __attribute__((ext_vector_type(2)));
typedef float v8f __attribute__((ext_vector_type(8)));

// ---------------- elementwise helpers ----------------

__global__ void fill_f32(float* __restrict__ p, float v, int n) {
    int i = blockIdx.x * blockDim.x + threadIdx.x;
    if (i < n) p[i] = v;
}

__global__ void deg_count(const int* __restrict__ dst, float* __restrict__ deg, int E) {
    int e = blockIdx.x * blockDim.x + threadIdx.x;
    if (e < E) atomicAdd(&deg[dst[e]], 1.0f);
}

__global__ void rsqrt_inplace(float* __restrict__ p, int n) {
    int i = blockIdx.x * blockDim.x + threadIdx.x;
    if (i < n) p[i] = rsqrtf(p[i]);
}

__global__ void relu_inplace(float* __restrict__ p, int n) {
    int i = blockIdx.x * blockDim.x + threadIdx.x;
    if (i < n) p[i] = fmaxf(p[i], 0.0f);
}

// out[i,f] = 0.5*(ba[f]+bb[f]) + 0.5*dinvA[i]^2*xwa[i,f] + 0.5*dinvB[i]^2*xwb[i,f]
// (bias mean + self-loop contributions of both relations; edge contributions are
//  atomically added afterwards by scatter_edges.)
__global__ void init_accum(const float* __restrict__ xwa, const float* __restrict__ xwb,
                           const float* __restrict__ dinvA, const float* __restrict__ dinvB,
                           const float* __restrict__ ba, const float* __restrict__ bb,
                           float* __restrict__ out, int Nn, int F) {
    int i = blockIdx.x * blockDim.x + threadIdx.x;
    int node = i / F;
    int f = i - node * F;
    if (node < Nn) {
        float sa = dinvA[node];
        float sb = dinvB[node];
        out[i] = 0.5f * (ba[f] + bb[f])
               + 0.5f * sa * sa * xwa[i]
               + 0.5f * sb * sb * xwb[i];
    }
}

// Per edge: out[dst] += 0.5 * dinv[src]*dinv[dst] * xw[src]  (float4-vectorized gather,
// per-float global_atomic_add_f32 scatter). This is the HBM/atomic-bound hot loop.
__global__ void scatter_edges(const int* __restrict__ ei,   // [2,E] flat: src=ei[e], dst=ei[E+e]
                              const float* __restrict__ xw,
                              const float* __restrict__ dinv,
                              float* __restrict__ out,
                              int E, int F) {
    const int cpe = F >> 2;                         // float4 chunks per edge
    long long tid = (long long)blockIdx.x * blockDim.x + threadIdx.x;
    long long total = (long long)E * cpe;
    if (tid >= total) return;
    int e = (int)(tid / cpe);
    int c = (int)(tid - (long long)e * cpe);
    int s = ei[e];
    int d = ei[E + e];
    float w = 0.5f * dinv[s] * dinv[d];
    const float4 v = *(const float4*)(xw + (long long)s * F + c * 4);
    float* o = out + (long long)d * F + c * 4;
    atomicAdd(o + 0, w * v.x);
    atomicAdd(o + 1, w * v.y);
    atomicAdd(o + 2, w * v.z);
    atomicAdd(o + 3, w * v.w);
}

// ---------------- WMMA fp32 GEMM: Y[M, NT*16] = X[M, 64] @ W[64, NT*16] ----------------
// K is fixed at 64 for both layers. W is staged in LDS once per block; each of the
// 8 waves in a block produces a 16-row x (NT*16)-col strip with NT accumulators,
// reusing each A fragment across NT v_wmma_f32_16x16x4_f32 issues.
//
// ISA VGPR layouts (CDNA5 ISA):
//   A 16x4 f32 : lane l<16 -> M=l, K={k,k+1}; lane l>=16 -> M=l-16, K={k+2,k+3}
//   B 4x16 f32 : symmetric, N = lane&15, same K-pair split
//   C/D 16x16  : VGPR v -> row v (lanes 0-15) / row v+8 (lanes 16-31), col = lane&15
template <int NT>
__global__ void wmma_gemm_f32(const float* __restrict__ X, const float* __restrict__ W,
                              float* __restrict__ Y, int M) {
    constexpr int K = 64;
    constexpr int Nout = NT * 16;
    __shared__ float wlds[K * Nout];

    // Stage W into LDS (one pass, all threads), then barrier.
    for (int i = threadIdx.x; i < K * Nout; i += blockDim.x)
        wlds[i] = W[i];
    __syncthreads();

    const int waves_per_block = blockDim.x >> 5;
    const int mt = blockIdx.x * waves_per_block + (threadIdx.x >> 5);
    const int m0 = mt * 16;
    if (m0 >= M) return;                    // wave-uniform: EXEC all-ones at WMMA

    const int lane = threadIdx.x & 31;
    const int half = lane >> 4;             // 0 | 1
    const int l15  = lane & 15;
    const long long arow = m0 + l15;        // A row for this lane

    v8f acc[NT];
#pragma unroll
    for (int t = 0; t < NT; ++t) acc[t] = (v8f){};

#pragma unroll 4
    for (int k = 0; k < K; k += 4) {
        const int kk = k + (half << 1);
        const v2f a = *(const v2f*)(X + arow * K + kk);   // 64-bit load, K-contiguous
#pragma unroll
        for (int t = 0; t < NT; ++t) {
            const int bcol = t * 16 + l15;
            v2f b = { wlds[kk * Nout + bcol], wlds[(kk + 1) * Nout + bcol] };
            acc[t] = __builtin_amdgcn_wmma_f32_16x16x4_f32(
                /*neg_a=*/false, a, /*neg_b=*/false, b,
                /*c_mod=*/(short)0, acc[t], /*reuse_a=*/false, /*reuse_b=*/false);
        }
    }

    const long long r0 = m0 + half * 8;
#pragma unroll
    for (int t = 0; t < NT; ++t) {
#pragma unroll
        for (int v = 0; v < 8; ++v) {
            long long r = r0 + v;
            if (r < M) Y[r * Nout + t * 16 + l15] = acc[t][v];
        }
    }
}

// ---------------- launch ----------------

static inline int ceil_div(long long a, int b) { return (int)((a + b - 1) / b); }

extern "C" void kernel_launch(void* const* d_in, const int* in_sizes, int n_in,
                              void* d_out, int out_size, void* d_ws, size_t ws_size,
                              hipStream_t stream) {
    const float* x    = (const float*)d_in[0];
    const int*   eiA  = (const int*)  d_in[1];
    const int*   eiB  = (const int*)  d_in[2];
    const float* W1a  = (const float*)d_in[3];
    const float* b1a  = (const float*)d_in[4];
    const float* W1b  = (const float*)d_in[5];
    const float* b1b  = (const float*)d_in[6];
    const float* W2a  = (const float*)d_in[7];
    const float* b2a  = (const float*)d_in[8];
    const float* W2b  = (const float*)d_in[9];
    const float* b2b  = (const float*)d_in[10];
    float* out = (float*)d_out;

    const int IN_F = 64, HID = 64, OUT_C = 16;
    const int Nn = in_sizes[0] / IN_F;
    const int E  = in_sizes[1] / 2;
    (void)IN_F;

    float* ws    = (float*)d_ws;
    float* dinvA = ws;                          // Nn  (holds deg, then rsqrt'd in place)
    float* dinvB = ws + Nn;                     // Nn
    float* xw1a  = ws + 2 * (size_t)Nn;         // Nn*64
    float* xw1b  = xw1a + (size_t)Nn * HID;     // Nn*64
    float* h     = xw1b + (size_t)Nn * HID;     // Nn*64
    float* y2a   = xw1a;                        // reuse after layer-1 scatter completes
    float* y2b   = xw1b;

    const int B = 256;
    const int WPB = B / 32;                     // 8 row-tiles per block

    // --- degrees -> dinv (shared by both layers; deg starts at 1 for self-loop) ---
    fill_f32<<<ceil_div(2LL * Nn, B), B, 0, stream>>>(dinvA, 1.0f, 2 * Nn);
    deg_count<<<ceil_div(E, B), B, 0, stream>>>(eiA + E, dinvA, E);
    deg_count<<<ceil_div(E, B), B, 0, stream>>>(eiB + E, dinvB, E);
    rsqrt_inplace<<<ceil_div(2LL * Nn, B), B, 0, stream>>>(dinvA, 2 * Nn);

    // --- layer 1 GEMMs (WMMA, NT=4 -> 64-wide strip per wave): xw = x @ W1{a,b} ---
    {
        int mtiles = (Nn + 15) / 16;
        int blocks = ceil_div(mtiles, WPB);
        wmma_gemm_f32<4><<<blocks, B, 0, stream>>>(x, W1a, xw1a, Nn);
        wmma_gemm_f32<4><<<blocks, B, 0, stream>>>(x, W1b, xw1b, Nn);
    }

    // --- layer 1 aggregate: bias + self-loops, then edge scatter (both relations) ---
    init_accum<<<ceil_div((long long)Nn * HID, B), B, 0, stream>>>(
        xw1a, xw1b, dinvA, dinvB, b1a, b1b, h, Nn, HID);
    {
        long long thr = (long long)E * (HID / 4);
        scatter_edges<<<ceil_div(thr, B), B, 0, stream>>>(eiA, xw1a, dinvA, h, E, HID);
        scatter_edges<<<ceil_div(thr, B), B, 0, stream>>>(eiB, xw1b, dinvB, h, E, HID);
    }
    relu_inplace<<<ceil_div((long long)Nn * HID, B), B, 0, stream>>>(h, Nn * HID);

    // --- layer 2 GEMMs (WMMA, NT=1): y2 = h @ W2{a,b} ---
    {
        int mtiles = (Nn + 15) / 16;
        int blocks = ceil_div(mtiles, WPB);
        wmma_gemm_f32<1><<<blocks, B, 0, stream>>>(h, W2a, y2a, Nn);
        wmma_gemm_f32<1><<<blocks, B, 0, stream>>>(h, W2b, y2b, Nn);
    }

    // --- layer 2 aggregate into d_out ---
    init_accum<<<ceil_div((long long)Nn * OUT_C, B), B, 0, stream>>>(
        y2a, y2b, dinvA, dinvB, b2a, b2b, out, Nn, OUT_C);
    {
        long long thr = (long long)E * (OUT_C / 4);
        scatter_edges<<<ceil_div(thr, B), B, 0, stream>>>(eiA, y2a, dinvA, out, E, OUT_C);
        scatter_edges<<<ceil_div(thr, B), B, 0, stream>>>(eiB, y2b, dinvB, out, E, OUT_C);
    }
}